// RayMarcher_32375463477277
// MI455X (gfx1250) — compile-verified
//
#include <hip/hip_runtime.h>
#include <hip/hip_bf16.h>

// ---------------------------------------------------------------------------
// MVP-style volumetric ray marcher for MI455X (gfx1250, wave32).
//  - fp16 template (64 prims x 8^3 voxels x RGBA = 256KB) resident in LDS
//    (320KB/WGP), filled by one Tensor Data Mover descriptor (TENSORcnt).
//  - camera rotation of ray directions done with V_WMMA_F32_16X16X4_F32.
//  - march loop: 57 steps x 64 prims, order-preserving front-to-back blend.
// ---------------------------------------------------------------------------

#define IMAGE_H   128
#define IMAGE_W   128
#define NRAYS     (IMAGE_H * IMAGE_W)
#define KP        64
#define NVOX      512            // 8^3
#define VOLR      64.0f
#define DTN       0.0625f        // 4/64
#define NSTEPS    57             // ceil(2*sqrt(3)/DTN)+1
#define TERMTH    0.99f

typedef float      v2f __attribute__((ext_vector_type(2)));
typedef float      v8f __attribute__((ext_vector_type(8)));
typedef _Float16   h4  __attribute__((ext_vector_type(4)));
typedef unsigned int u4v __attribute__((ext_vector_type(4)));
typedef int        i8v __attribute__((ext_vector_type(8)));
typedef int        i4v __attribute__((ext_vector_type(4)));

#if __has_builtin(__builtin_amdgcn_tensor_load_to_lds)
#define HAS_TDM 1
#else
#define HAS_TDM 0
#endif
#if __has_builtin(__builtin_amdgcn_wmma_f32_16x16x4_f32)
#define HAS_WMMA4 1
#else
#define HAS_WMMA4 0
#endif

// Dynamic LDS layout (template MUST be at offset 0: TDM lds_addr field = 0)
#define SM_TMPL_OFF   0                         // 64*512*4 halfs = 262144 B
#define SM_CONST_OFF  262144                    // 64 prims * 12 floats = 3072 B
#define SM_SCR_OFF    (SM_CONST_OFF + 3072)     // 256 rays * 2 floats  = 2048 B
#define SM_DIR_OFF    (SM_SCR_OFF + 2048)       // 256 rays * 3 floats  = 3072 B
#define SMEM_BYTES    (SM_DIR_OFF + 3072)       // 270336 B  (< 320KB WGP LDS)

// ---------------------------------------------------------------------------
// Kernel 1: convert template (B,K,4,8,8,8) f32 channel-major -> voxel-major
//           packed fp16 RGBA (8 bytes / voxel) in workspace.
// ---------------------------------------------------------------------------
__global__ void pack_tmpl_f16(const float* __restrict__ rgba,
                              h4* __restrict__ out) {
    int idx = blockIdx.x * blockDim.x + threadIdx.x;   // (b*K + k)*512 + v
    int v   = idx & (NVOX - 1);
    int bk  = idx >> 9;
    const float* src = rgba + (size_t)bk * (4 * NVOX) + v;
    h4 o;
    o.x = (_Float16)src[0 * NVOX];
    o.y = (_Float16)src[1 * NVOX];
    o.z = (_Float16)src[2 * NVOX];
    o.w = (_Float16)src[3 * NVOX];
    out[idx] = o;
}

// ---------------------------------------------------------------------------
// Kernel 2: ray marcher. 256 threads = 256 rays, grid (64, B).
// ---------------------------------------------------------------------------
__launch_bounds__(256, 1)
__global__ void march_kernel(const float* __restrict__ prim_pos,
                             const float* __restrict__ prim_rot,
                             const float* __restrict__ prim_scale,
                             const float* __restrict__ Kmat,
                             const float* __restrict__ RT,
                             const h4*    __restrict__ tmpl16,
                             float*       __restrict__ out) {
    extern __shared__ char smem[];
    _Float16* sT   = (_Float16*)(smem + SM_TMPL_OFF);
    float*    sC   = (float*)(smem + SM_CONST_OFF);
    float*    sScr = (float*)(smem + SM_SCR_OFF);
    float*    sDir = (float*)(smem + SM_DIR_OFF);

    const int tid = threadIdx.x;
    const int b   = blockIdx.y;
    const int ray = blockIdx.x * 256 + tid;

    // ---- template -> LDS -------------------------------------------------
#if HAS_TDM
    if (tid < 32) {   // wave 0 issues one TDM descriptor: 32768 x 8B -> LDS 0
        unsigned long long ga =
            (unsigned long long)(const void*)(tmpl16 + (size_t)b * KP * NVOX);
        u4v g0 = { 1u,                                   // count=1 (valid D#)
                   (unsigned)SM_TMPL_OFF,                // lds_addr
                   (unsigned)(ga & 0xFFFFFFFFu),         // global_addr[31:0]
                   (unsigned)((ga >> 32) & 0x01FFFFFFu) | (2u << 30) }; // type=2
        i8v g1 = { (int)(3u << 16),        // data_size = 8 bytes
                   (int)0x80000000u,       // tensor_dim0[15:0]=0x8000 (hi half)
                   (int)0x00010000u,       // tensor_dim0 hi=0, tensor_dim1=1
                   (int)0x80000000u,       // tile_dim0 = 0x8000
                   1,                      // tile_dim1=1, tile_dim2=0
                   32768,                  // tensor_dim0_stride lo
                   0, 0 };
        i4v g2 = { 1, 1, 0, 0 };           // tensor_dim2=1, tensor_dim3=1
        i4v g3 = { 0, 0, 0, 0 };
        i8v g4 = { 0, 0, 0, 0, 0, 0, 0, 0 };   // unused trailing group (2D tensor)
        __builtin_amdgcn_tensor_load_to_lds(g0, g1, g2, g3, g4, 0);
    }
#else
    {   // cooperative fallback copy (1KB per thread)
        const u4v* src = (const u4v*)(tmpl16 + (size_t)b * KP * NVOX);
        u4v*       dst = (u4v*)sT;
        for (int i = tid; i < (KP * NVOX * 8) / 16; i += 256) dst[i] = src[i];
    }
#endif

    // ---- per-prim constants: Rs = scale .* R^T,  bias = -Rs*ppos ---------
    if (tid < KP) {
        const int k = tid;
        const float* R = prim_rot   + ((size_t)b * KP + k) * 9;
        const float* P = prim_pos   + ((size_t)b * KP + k) * 3;
        const float* S = prim_scale + ((size_t)b * KP + k) * 3;
        float ppx = P[0] * (1.0f / VOLR);
        float ppy = P[1] * (1.0f / VOLR);
        float ppz = P[2] * (1.0f / VOLR);
#pragma unroll
        for (int i = 0; i < 3; ++i) {
            float s  = S[i];
            float r0 = s * R[0 * 3 + i];
            float r1 = s * R[1 * 3 + i];
            float r2 = s * R[2 * 3 + i];
            sC[k * 12 + i * 4 + 0] = r0;
            sC[k * 12 + i * 4 + 1] = r1;
            sC[k * 12 + i * 4 + 2] = r2;
            sC[k * 12 + i * 4 + 3] = -(r0 * ppx + r1 * ppy + r2 * ppz);
        }
    }

    // ---- camera setup ----------------------------------------------------
    const float* rt = RT + b * 12;            // rows of [R_w2c | t]
    float t0 = rt[3], t1 = rt[7], t2 = rt[11];
    float cpx = -(rt[0] * t0 + rt[4] * t1 + rt[8]  * t2);
    float cpy = -(rt[1] * t0 + rt[5] * t1 + rt[9]  * t2);
    float cpz = -(rt[2] * t0 + rt[6] * t1 + rt[10] * t2);
    const float* Kb = Kmat + b * 9;
    const int pxI = ray & (IMAGE_W - 1);
    const int pyI = ray >> 7;
    float dx = ((float)pxI - Kb[2]) / Kb[0];
    float dy = ((float)pyI - Kb[5]) / Kb[4];
    sScr[tid * 2 + 0] = dx;
    sScr[tid * 2 + 1] = dy;

    __syncthreads();   // sC + sScr visible

    // ---- raydir = R_w2c^T * (dx,dy,1) via WMMA 16x16x4 F32 ---------------
#if HAS_WMMA4
    {
        const int wave = tid >> 5, lane = tid & 31;
        const int half = lane >> 4, lm = lane & 15;
        // B (4x16): row K = R_w2c row K (cols 0..2), row 3 = 0.
        // Layout: VGPR0 lanes0-15 K=0 / lanes16-31 K=1; VGPR1 K=2 / K=3.
        float b0 = 0.f, b1 = 0.f;
        if (lm < 3) {
            b0 = half ? rt[1 * 4 + lm] : rt[0 * 4 + lm];
            b1 = half ? 0.f            : rt[2 * 4 + lm];
        }
        v2f Bv; Bv.x = b0; Bv.y = b1;
#pragma unroll
        for (int tI = 0; tI < 2; ++tI) {          // two 16-ray tiles per wave
            int rbase = wave * 32 + tI * 16;
            // A (16x4): row r = (dx, dy, 1, 0).  lanes0-15: K0,K1; 16-31: K2,K3
            float a0 = half ? 1.0f : sScr[(rbase + lm) * 2 + 0];
            float a1 = half ? 0.0f : sScr[(rbase + lm) * 2 + 1];
            v2f Av; Av.x = a0; Av.y = a1;
            v8f Cv = {0.f, 0.f, 0.f, 0.f, 0.f, 0.f, 0.f, 0.f};
            v8f Dv = __builtin_amdgcn_wmma_f32_16x16x4_f32(
                false, Av, false, Bv, (short)0, Cv, false, false);
            if (lm < 3) {                          // D: lane owns column N=lm
#pragma unroll
                for (int v = 0; v < 8; ++v)
                    sDir[(rbase + v + half * 8) * 3 + lm] = Dv[v];
            }
        }
    }
#endif

#if HAS_TDM
    if (tid < 32) {
#if __has_builtin(__builtin_amdgcn_s_wait_tensorcnt)
        __builtin_amdgcn_s_wait_tensorcnt(0);
#else
        asm volatile("s_wait_tensorcnt 0x0" ::: "memory");
#endif
    }
#endif
    __syncthreads();   // template + sDir visible

    // ---- finalize ray ----------------------------------------------------
    float ddx, ddy, ddz;
#if HAS_WMMA4
    ddx = sDir[tid * 3 + 0];
    ddy = sDir[tid * 3 + 1];
    ddz = sDir[tid * 3 + 2];
#else
    ddx = rt[0] * dx + rt[4] * dy + rt[8];
    ddy = rt[1] * dx + rt[5] * dy + rt[9];
    ddz = rt[2] * dx + rt[6] * dy + rt[10];
#endif
    float inv = rsqrtf(ddx * ddx + ddy * ddy + ddz * ddz);
    ddx *= inv; ddy *= inv; ddz *= inv;

    const float rpx = cpx * (1.0f / VOLR);
    const float rpy = cpy * (1.0f / VOLR);
    const float rpz = cpz * (1.0f / VOLR);

    auto safe = [](float d) {
        return (fabsf(d) < 1e-8f) ? ((d >= 0.f) ? 1e-8f : -1e-8f) : d;
    };
    float sx = safe(ddx), sy = safe(ddy), sz = safe(ddz);
    float t1x = (-1.f - rpx) / sx, t2x = (1.f - rpx) / sx;
    float t1y = (-1.f - rpy) / sy, t2y = (1.f - rpy) / sy;
    float t1z = (-1.f - rpz) / sz, t2z = (1.f - rpz) / sz;
    float tmin = fmaxf(fmaxf(fminf(t1x, t2x), fminf(t1y, t2y)), fminf(t1z, t2z));
    tmin = fmaxf(tmin, 0.f);
    float tmax = fminf(fminf(fmaxf(t1x, t2x), fmaxf(t1y, t2y)), fmaxf(t1z, t2z));
    tmax = fmaxf(tmax, tmin);

    // ---- march -----------------------------------------------------------
    float alpha = 0.f, oR = 0.f, oG = 0.f, oB = 0.f;
    for (int s = 0; s < NSTEPS; ++s) {
        float t = fmaf((float)s, DTN, tmin);
        bool active = (t < tmax) && (alpha <= TERMTH);
        if (!__any(active)) break;                  // wave-wide termination
        float posx = fmaf(t, ddx, rpx);
        float posy = fmaf(t, ddy, rpy);
        float posz = fmaf(t, ddz, rpz);
        float cum = alpha;                          // raw cumsum within step
        for (int k = 0; k < KP; ++k) {
            const float* C = sC + k * 12;           // LDS broadcast reads
            float yx = fmaf(C[0], posx, fmaf(C[1], posy, fmaf(C[2],  posz, C[3])));
            float yy = fmaf(C[4], posx, fmaf(C[5], posy, fmaf(C[6],  posz, C[7])));
            float yz = fmaf(C[8], posx, fmaf(C[9], posy, fmaf(C[10], posz, C[11])));
            float ax = fabsf(yx), ay = fabsf(yy), az = fabsf(yz);
            bool inside = active && (ax < 1.f) && (ay < 1.f) && (az < 1.f);
            if (__any(inside)) {
                // trilinear sample from LDS fp16 template
                float ux = fmaf(yx, 4.f, 3.5f);
                float uy = fmaf(yy, 4.f, 3.5f);
                float uz = fmaf(yz, 4.f, 3.5f);
                float fx0 = floorf(ux), fy0 = floorf(uy), fz0 = floorf(uz);
                float wx1 = ux - fx0, wy1 = uy - fy0, wz1 = uz - fz0;
                float wxf[2] = {1.f - wx1, wx1};
                float wyf[2] = {1.f - wy1, wy1};
                float wzf[2] = {1.f - wz1, wz1};
                int ix0 = (int)fx0, iy0 = (int)fy0, iz0 = (int)fz0;
                int cxi[2] = { min(max(ix0, 0), 7), min(max(ix0 + 1, 0), 7) };
                int cyi[2] = { min(max(iy0, 0), 7), min(max(iy0 + 1, 0), 7) };
                int czi[2] = { min(max(iz0, 0), 7), min(max(iz0 + 1, 0), 7) };
                float smR = 0.f, smG = 0.f, smB = 0.f, smA = 0.f;
                const _Float16* Tk = sT + k * (NVOX * 4);
#pragma unroll
                for (int cz = 0; cz < 2; ++cz)
#pragma unroll
                    for (int cy = 0; cy < 2; ++cy) {
                        float wzy = wzf[cz] * wyf[cy];
#pragma unroll
                        for (int cx = 0; cx < 2; ++cx) {
                            int flat = (czi[cz] * 8 + cyi[cy]) * 8 + cxi[cx];
                            h4 hv = *(const h4*)(Tk + flat * 4); // ds_load_b64
                            float w = wzy * wxf[cx];
                            smR = fmaf((float)hv.x, w, smR);
                            smG = fmaf((float)hv.y, w, smG);
                            smB = fmaf((float)hv.z, w, smB);
                            smA = fmaf((float)hv.w, w, smA);
                        }
                    }
                float ax2 = ax * ax, ay2 = ay * ay, az2 = az * az;
                float ax4 = ax2 * ax2, ay4 = ay2 * ay2, az4 = az2 * az2;
                float s8 = ax4 * ax4 + ay4 * ay4 + az4 * az4;
                float fade = __expf(-8.f * s8);
                float a_k  = inside ? (smA * fade * DTN) : 0.f;
                float cumN = cum + a_k;
                float contrib = fminf(cumN, 1.f) - fminf(cum, 1.f);
                oR = fmaf(smR, contrib, oR);
                oG = fmaf(smG, contrib, oG);
                oB = fmaf(smB, contrib, oB);
                alpha += contrib;
                cum = cumN;
            }
        }
    }

    const size_t obase = (size_t)b * 4 * NRAYS + ray;
    out[obase + 0 * NRAYS] = oR;
    out[obase + 1 * NRAYS] = oG;
    out[obase + 2 * NRAYS] = oB;
    out[obase + 3 * NRAYS] = alpha;
}

// ---------------------------------------------------------------------------
extern "C" void kernel_launch(void* const* d_in, const int* in_sizes, int n_in,
                              void* d_out, int out_size, void* d_ws, size_t ws_size,
                              hipStream_t stream) {
    const float* prim_rgba  = (const float*)d_in[0];
    const float* prim_pos   = (const float*)d_in[1];
    const float* prim_rot   = (const float*)d_in[2];
    const float* prim_scale = (const float*)d_in[3];
    const float* Kmat       = (const float*)d_in[4];
    const float* RT         = (const float*)d_in[5];
    float* out = (float*)d_out;
    h4* tmpl16 = (h4*)d_ws;                 // 2*64*512*8B = 512KB scratch

    (void)hipFuncSetAttribute((const void*)march_kernel,
                              hipFuncAttributeMaxDynamicSharedMemorySize,
                              SMEM_BYTES);

    const int nvoxTot = 2 * KP * NVOX;      // 65536 voxels
    pack_tmpl_f16<<<nvoxTot / 256, 256, 0, stream>>>(prim_rgba, tmpl16);

    dim3 grid(NRAYS / 256, 2);
    march_kernel<<<grid, 256, SMEM_BYTES, stream>>>(
        prim_pos, prim_rot, prim_scale, Kmat, RT, tmpl16, out);
}